// SoftMoe_67113158967412
// MI455X (gfx1250) — compile-verified
//
#include <hip/hip_runtime.h>
#include <hip/hip_bf16.h>
#include <math.h>

// ---- problem constants (match reference) ----
#define BB   8
#define NN   4096
#define DD   768
#define EE   8
#define SS   2
#define ES   16        // E*S
#define FF   3072      // 4*D

typedef __attribute__((ext_vector_type(2))) float    v2f;
typedef __attribute__((ext_vector_type(8))) float    v8f;
typedef __attribute__((ext_vector_type(4))) unsigned v4u;
typedef __attribute__((ext_vector_type(8))) unsigned v8u;

// D = A(16x4) * B(4x16) + C, fp32 WMMA
__device__ __forceinline__ v8f wmma_f32_4(v2f a, v2f b, v8f c) {
    return __builtin_amdgcn_wmma_f32_16x16x4_f32(
        /*neg_a=*/false, a, /*neg_b=*/false, b,
        /*c_mod=*/(short)0, c, /*reuse_a=*/false, /*reuse_b=*/false);
}

// Async memory -> LDS mover (16B per lane), tracked by ASYNCcnt.
__device__ __forceinline__ void async_to_lds_b128(unsigned lds_off, const void* gptr) {
    asm volatile("global_load_async_to_lds_b128 %0, %1, off"
                 :: "v"(lds_off), "v"((unsigned long long)(uintptr_t)gptr)
                 : "memory");
}
__device__ __forceinline__ void wait_asynccnt0() {
    asm volatile("s_wait_asynccnt 0x0" ::: "memory");
}

// TDM: 1-D contiguous tile (nbytes, multiple of 4) from global -> LDS.
// D# per CDNA5 ISA. Uniform operands -> SGPR groups.
__device__ __forceinline__ void tdm_load_1d(unsigned lds_addr, const void* gptr, unsigned nfloats) {
    const unsigned long long ga = (unsigned long long)(uintptr_t)gptr;
    v4u g0;
    g0.x = 1u;                                   // count=1, user mode
    g0.y = lds_addr;                             // lds_addr [63:32]
    g0.z = (unsigned)(ga & 0xffffffffu);         // global_addr [95:64]
    g0.w = (unsigned)((ga >> 32) & 0x1ffffffu)   // global_addr [120:96]
         | (2u << 30);                           // type=2 ("image") [127:126]
    v8u g1;
    g1.s0 = (2u << 16);                          // data_size=4B, mask=0, no pad/iter
    g1.s1 = (nfloats & 0xffffu) << 16;           // tensor_dim0[15:0]
    g1.s2 = ((nfloats >> 16) & 0xffffu)          // tensor_dim0[31:16]
          | (1u << 16);                          // tensor_dim1 = 1
    g1.s3 = (nfloats & 0xffffu) << 16;           // tile_dim0 (<=65535)
    g1.s4 = 1u;                                  // tile_dim1=1, tile_dim2=0
    g1.s5 = nfloats;                             // tensor_dim0_stride[31:0]
    g1.s6 = 0u;                                  // dim0_stride hi, dim1_stride lo
    g1.s7 = 0u;                                  // dim1_stride hi
    asm volatile("tensor_load_to_lds %0, %1" :: "s"(g0), "s"(g1) : "memory");
}

// =====================================================================
// Kernel 1: logits = x @ phi ; dispatch = softmax(logits, axis=-1)
// grid 256 x block 256 (8 waves); each wave: one 16-token tile.
// phi (48 KB) staged into LDS with async movers.
// =====================================================================
__global__ void k_logits_dispatch(const float* __restrict__ x,
                                  const float* __restrict__ phi,
                                  float* __restrict__ logits,
                                  float* __restrict__ dispatch) {
    __shared__ __align__(16) float phis[DD * ES];           // 48 KB
    const int t = threadIdx.x;
    {
        const unsigned lds_base = (unsigned)(uintptr_t)phis;
        for (int i = t; i < (DD * ES) / 4; i += 256)        // 12 x b128 per lane
            async_to_lds_b128(lds_base + i * 16, phi + i * 4);
        wait_asynccnt0();
    }
    __syncthreads();

    const int wave = t >> 5, lane = t & 31;
    const int l16 = lane & 15, half = lane >> 4;
    const long tile = (long)blockIdx.x * 8 + wave;  // over B*N/16 tiles
    const long t0 = tile * 16;
    const float* xrow = x + (t0 + l16) * DD;        // stays resident in 192MB L2 for k3

    v8f c = {};
    #pragma unroll 8
    for (int k0 = 0; k0 < DD; k0 += 4) {
        const int k = k0 + 2 * half;
        v2f a, bm;
        a.x = xrow[k];
        a.y = xrow[k + 1];
        bm.x = phis[k * ES + l16];
        bm.y = phis[(k + 1) * ES + l16];
        c = wmma_f32_4(a, bm, c);
    }
    // c[r]: token = t0 + r + 8*half, slot = l16. Row softmax across lanes.
    #pragma unroll
    for (int r = 0; r < 8; ++r) {
        float v = c[r];
        float m = v;
        #pragma unroll
        for (int d = 1; d < 16; d <<= 1) m = fmaxf(m, __shfl_xor(m, d, 32));
        float e = expf(v - m);
        float s = e;
        #pragma unroll
        for (int d = 1; d < 16; d <<= 1) s += __shfl_xor(s, d, 32);
        const long tok = t0 + r + 8 * half;
        logits[tok * ES + l16]   = v;
        dispatch[tok * ES + l16] = e / s;
    }
}

// =====================================================================
// Kernel 2: per-(b,slot) online logsumexp over tokens (combine stats)
// grid 8 x block 256.
// =====================================================================
__global__ void k_colstats(const float* __restrict__ logits,
                           float* __restrict__ colmax,
                           float* __restrict__ colsum) {
    const int b = blockIdx.x, t = threadIdx.x;
    const int slot = t & 15, grp = t >> 4;           // 16 token-groups
    float m = -INFINITY, s = 0.f;
    for (int n = grp; n < NN; n += 16) {
        const float v = logits[((long)b * NN + n) * ES + slot];
        const float mn = fmaxf(m, v);
        s = s * expf(m - mn) + expf(v - mn);
        m = mn;
    }
    __shared__ float sm[256], ss[256];
    sm[t] = m; ss[t] = s;
    __syncthreads();
    for (int stride = 128; stride >= 16; stride >>= 1) {
        if (t < stride) {
            const float m2 = sm[t + stride], s2 = ss[t + stride];
            const float mn = fmaxf(sm[t], m2);
            ss[t] = ss[t] * expf(sm[t] - mn) + s2 * expf(m2 - mn);
            sm[t] = mn;
        }
        __syncthreads();
    }
    if (t < 16) { colmax[b * ES + t] = sm[t]; colsum[b * ES + t] = ss[t]; }
}

// =====================================================================
// Kernel 3: xs[b, slot, d] = sum_n dispatch[b,n,slot] * x[b,n,d]
// M=16 slots, N=16-dim tile, K=4096 tokens. grid 96 x block 128.
// =====================================================================
__global__ void k_xs(const float* __restrict__ x,
                     const float* __restrict__ dispatch,
                     float* __restrict__ xsbuf) {
    const int t = threadIdx.x;
    const int wid = blockIdx.x * 4 + (t >> 5);       // 0..383
    const int lane = t & 31, l16 = lane & 15, half = lane >> 4;
    const int b = wid / 48, dt = wid % 48;
    const int d0 = dt * 16;
    const float* xb = x + (long)b * NN * DD;
    const float* db = dispatch + (long)b * NN * ES;

    v8f c = {};
    #pragma unroll 4
    for (int n0 = 0; n0 < NN; n0 += 4) {
        const int k = n0 + 2 * half;
        __builtin_prefetch(&xb[(long)(k + 64) * DD + d0 + l16], 0, 0);
        v2f a, bm;
        a.x = db[k * ES + l16];                       // A[slot=l16][tok=k]
        a.y = db[(k + 1) * ES + l16];
        bm.x = xb[(long)k * DD + d0 + l16];           // B[tok=k][d]
        bm.y = xb[(long)(k + 1) * DD + d0 + l16];
        c = wmma_f32_4(a, bm, c);
    }
    #pragma unroll
    for (int r = 0; r < 8; ++r) {
        const int slot = r + 8 * half;
        xsbuf[((long)b * ES + slot) * DD + d0 + l16] = c[r];
    }
}

// =====================================================================
// Kernel 4a: h[e] = gelu( xs_rows[e] @ w1[e] + b1[e] ), rows=(b,s)=16
// grid 384 x block 128. w1 streamed once -> non-temporal loads.
// =====================================================================
__global__ void k_ffn1(const float* __restrict__ xsbuf,
                       const float* __restrict__ w1,
                       const float* __restrict__ b1,
                       float* __restrict__ hbuf) {
    const int t = threadIdx.x;
    const int wid = blockIdx.x * 4 + (t >> 5);       // 0..1535
    const int lane = t & 31, l16 = lane & 15, half = lane >> 4;
    const int e = wid / 192, ft = wid % 192;
    const int f0 = ft * 16;
    // A row l16 = (b,s): b=l16>>1, s=l16&1 -> xsbuf[b, e*2+s, :]
    const float* arow = xsbuf + ((long)(l16 >> 1) * ES + e * SS + (l16 & 1)) * DD;
    const float* w1e = w1 + (long)e * DD * FF;

    v8f c = {};
    #pragma unroll 8
    for (int k0 = 0; k0 < DD; k0 += 4) {
        const int k = k0 + 2 * half;
        __builtin_prefetch(&w1e[(long)(k + 64) * FF + f0 + l16], 0, 0);
        v2f a, bm;
        a.x = arow[k];
        a.y = arow[k + 1];
        bm.x = __builtin_nontemporal_load(&w1e[(long)k * FF + f0 + l16]);
        bm.y = __builtin_nontemporal_load(&w1e[(long)(k + 1) * FF + f0 + l16]);
        c = wmma_f32_4(a, bm, c);
    }
    const float bias = b1[e * FF + f0 + l16];
    #pragma unroll
    for (int r = 0; r < 8; ++r) {
        float v = c[r] + bias;
        v = 0.5f * v * (1.0f + erff(v * 0.70710678118654752f));   // exact GELU
        const int row = r + 8 * half;
        hbuf[((long)e * 16 + row) * FF + f0 + l16] = v;
    }
}

// =====================================================================
// Kernel 4b: ys = h @ w2 + b2, scattered to ysbuf[b, e*S+s, d]
// grid 96 x block 128. w2 streamed once -> non-temporal loads.
// =====================================================================
__global__ void k_ffn2(const float* __restrict__ hbuf,
                       const float* __restrict__ w2,
                       const float* __restrict__ b2,
                       float* __restrict__ ysbuf) {
    const int t = threadIdx.x;
    const int wid = blockIdx.x * 4 + (t >> 5);       // 0..383
    const int lane = t & 31, l16 = lane & 15, half = lane >> 4;
    const int e = wid / 48, dt = wid % 48;
    const int d0 = dt * 16;
    const float* arow = hbuf + ((long)e * 16 + l16) * FF;
    const float* w2e = w2 + (long)e * FF * DD;

    v8f c = {};
    #pragma unroll 4
    for (int k0 = 0; k0 < FF; k0 += 4) {
        const int k = k0 + 2 * half;
        __builtin_prefetch(&w2e[(long)(k + 64) * DD + d0 + l16], 0, 0);
        v2f a, bm;
        a.x = arow[k];
        a.y = arow[k + 1];
        bm.x = __builtin_nontemporal_load(&w2e[(long)k * DD + d0 + l16]);
        bm.y = __builtin_nontemporal_load(&w2e[(long)(k + 1) * DD + d0 + l16]);
        c = wmma_f32_4(a, bm, c);
    }
    const float bias = b2[e * DD + d0 + l16];
    #pragma unroll
    for (int r = 0; r < 8; ++r) {
        const int row = r + 8 * half;                 // row = b*S + s
        const int bb = row >> 1, s = row & 1;
        ysbuf[((long)bb * ES + e * SS + s) * DD + d0 + l16] = c[r] + bias;
    }
}

// =====================================================================
// Kernel 5: y[b,n,d] = sum_slot combine[b,n,slot] * ys[b,slot,d]
// ys[b] (48 KB) staged to LDS by ONE Tensor-Data-Mover op (wave 0),
// TENSORcnt-waited. y written non-temporally (never re-read).
// grid 256 x block 256 (32 blocks per b, 8 token-tiles per block).
// =====================================================================
__global__ void k_combine(const float* __restrict__ logits,
                          const float* __restrict__ colmax,
                          const float* __restrict__ colsum,
                          const float* __restrict__ ysbuf,
                          float* __restrict__ y) {
    __shared__ __align__(16) float ys_lds[ES * DD];  // 48 KB
    __shared__ float cm[ES], cs[ES];
    const int b = blockIdx.x >> 5;
    const int blk = blockIdx.x & 31;
    const int t = threadIdx.x;
    if ((t >> 5) == 0) {                             // wave 0 drives the TDM
        tdm_load_1d((unsigned)(uintptr_t)ys_lds, ysbuf + (long)b * ES * DD, ES * DD);
        __builtin_amdgcn_s_wait_tensorcnt(0);
    }
    if (t < ES) { cm[t] = colmax[b * ES + t]; cs[t] = colsum[b * ES + t]; }
    __syncthreads();

    const int wave = t >> 5, lane = t & 31;
    const int l16 = lane & 15, half = lane >> 4;
    const long n0 = (long)blk * 128 + wave * 16;     // token tile within b

    // A tile: combine[tok = n0+l16][slot]; lanes<16 hold K={0,1}, lanes>=16 K={2,3}
    const float* lrow = logits + ((long)b * NN + n0 + l16) * ES;
    v2f A[4];
    #pragma unroll
    for (int kk = 0; kk < 4; ++kk) {
        const int slot = kk * 4 + 2 * half;
        A[kk].x = expf(lrow[slot]     - cm[slot])     / cs[slot];
        A[kk].y = expf(lrow[slot + 1] - cm[slot + 1]) / cs[slot + 1];
    }

    for (int dt = 0; dt < 48; ++dt) {
        const int d0 = dt * 16;
        v8f c = {};
        #pragma unroll
        for (int kk = 0; kk < 4; ++kk) {
            const int slot = kk * 4 + 2 * half;
            v2f bm;
            bm.x = ys_lds[slot * DD + d0 + l16];
            bm.y = ys_lds[(slot + 1) * DD + d0 + l16];
            c = wmma_f32_4(A[kk], bm, c);
        }
        #pragma unroll
        for (int r = 0; r < 8; ++r) {
            __builtin_nontemporal_store(
                c[r], &y[((long)b * NN + n0 + r + 8 * half) * DD + d0 + l16]);
        }
    }
}

// =====================================================================
extern "C" void kernel_launch(void* const* d_in, const int* in_sizes, int n_in,
                              void* d_out, int out_size, void* d_ws, size_t ws_size,
                              hipStream_t stream) {
    const float* x   = (const float*)d_in[0];   // [B,N,D]
    const float* phi = (const float*)d_in[1];   // [D,E*S]
    const float* w1  = (const float*)d_in[2];   // [E,D,4D]
    const float* b1  = (const float*)d_in[3];   // [E,4D]
    const float* w2  = (const float*)d_in[4];   // [E,4D,D]
    const float* b2  = (const float*)d_in[5];   // [E,D]
    float* y = (float*)d_out;                   // [B,N,D]

    float* ws = (float*)d_ws;
    float* logits   = ws;                         ws += (long)BB * NN * ES;
    float* dispatch = ws;                         ws += (long)BB * NN * ES;
    float* colmax   = ws;                         ws += BB * ES;
    float* colsum   = ws;                         ws += BB * ES;
    float* xsbuf    = ws;                         ws += (long)BB * ES * DD;
    float* hbuf     = ws;                         ws += (long)EE * 16 * FF;
    float* ysbuf    = ws;                         ws += (long)BB * ES * DD;
    (void)ws_size; (void)in_sizes; (void)n_in; (void)out_size;

    k_logits_dispatch<<<256, 256, 0, stream>>>(x, phi, logits, dispatch);
    k_colstats      <<<BB,  256, 0, stream>>>(logits, colmax, colsum);
    k_xs            <<<96,  128, 0, stream>>>(x, dispatch, xsbuf);
    k_ffn1          <<<384, 128, 0, stream>>>(xsbuf, w1, b1, hbuf);
    k_ffn2          <<<96,  128, 0, stream>>>(hbuf, w2, b2, ysbuf);
    k_combine       <<<256, 256, 0, stream>>>(logits, colmax, colsum, ysbuf, y);
}